// SenseEmbedding_48172353191982
// MI455X (gfx1250) — compile-verified
//
#include <hip/hip_runtime.h>
#include <math.h>

// Problem constants (from the reference)
#define VOCAB   100000
#define DIM     300
#define NSENSE  5
#define ILEN    12
#define BATCH   16384
#define SENSE_F (NSENSE * DIM)   // 1500 floats = 6000 bytes, contiguous per vocab entry
#define NCHUNK4 (DIM / 4)        // 75 float4 chunks per embedding row (300 = 75*16B)

#define WAVES_PER_BLOCK 4
#define BLOCK_THREADS   (WAVES_PER_BLOCK * 32)

typedef __attribute__((ext_vector_type(4))) unsigned int u32x4;
typedef __attribute__((ext_vector_type(8))) unsigned int u32x8;
typedef __attribute__((ext_vector_type(4))) float f32x4;

__device__ __forceinline__ f32x4 v4add(f32x4 a, f32x4 b) {
    f32x4 r; r.x = a.x + b.x; r.y = a.y + b.y; r.z = a.z + b.z; r.w = a.w + b.w;
    return r;
}
__device__ __forceinline__ float v4dot(f32x4 a, f32x4 b, float p) {
    p = fmaf(a.x, b.x, p);
    p = fmaf(a.y, b.y, p);
    p = fmaf(a.z, b.z, p);
    p = fmaf(a.w, b.w, p);
    return p;
}

__global__ __launch_bounds__(BLOCK_THREADS)
void SenseEmbedding_kernel(const int* __restrict__ x,
                           const float* __restrict__ Wg,
                           const float* __restrict__ Ws,
                           float* __restrict__ out)
{
    // One wave32 per batch row. Each wave gets a private 6KB LDS slice that the
    // Tensor Data Mover fills with W_s[x0] (5x300 f32) while the wave gathers
    // and sums the 10 context embeddings from W_g (L2-resident: 120MB < 192MB L2).
    __shared__ float smem[WAVES_PER_BLOCK][SENSE_F];

    const int lane = threadIdx.x & 31;
    const int wave = __builtin_amdgcn_readfirstlane((int)(threadIdx.x >> 5)); // uniform
    const int row  = (int)blockIdx.x * WAVES_PER_BLOCK + wave;                // uniform

    const int* __restrict__ xr = x + (size_t)row * ILEN;
    const int idx0 = xr[0];   // uniform (scalar load)
    const int idx1 = xr[1];   // uniform

    // ---------------- TDM: DMA W_s[idx0] -> LDS (1-D tile of 1500 dwords) ----
    const unsigned long long gaddr =
        (unsigned long long)(const void*)(Ws + (size_t)idx0 * SENSE_F);
    const unsigned int laddr =
        (unsigned int)(unsigned long long)(const void*)&smem[wave][0];

    // D# group 0 (128b): count=1 | lds_addr | global_addr[56:0] | type=2
    u32x4 g0;
    g0.x = 1u;                                            // count=1, user descriptor
    g0.y = laddr;                                         // lds_addr (bytes)
    g0.z = (unsigned int)gaddr;                           // global_addr[31:0]
    g0.w = ((unsigned int)(gaddr >> 32) & 0x01FFFFFFu)    // global_addr[56:32]
           | (2u << 30);                                  // type = 2 ("image")

    // D# group 1 (256b): data_size=4B, tensor_dim0=tile_dim0=1500, 1-D tile
    u32x8 g1;
    g1.s0 = (2u << 16);                 // workgroup_mask=0, data_size=2 (4 bytes)
    g1.s1 = ((unsigned)SENSE_F) << 16;  // tensor_dim0[15:0] @ bits 63:48
    g1.s2 = (1u << 16);                 // tensor_dim0 hi=0; tensor_dim1=1 @ bits 95:80
    g1.s3 = ((unsigned)SENSE_F) << 16;  // tile_dim0=1500 @ bits 127:112
    g1.s4 = 0u;                         // tile_dim1=0 (unused), tile_dim2=0
    g1.s5 = (unsigned)SENSE_F;          // tensor_dim0_stride lo32 = 1500
    g1.s6 = 0u;                         // stride0 hi16 = 0, stride1 lo16 = 0
    g1.s7 = 0u;

    // 2-group form (VADDR2/VADDR3 = NULL): tensors up to 2D. Tracked on TENSORcnt.
    asm volatile("tensor_load_to_lds %0, %1"
                 :
                 : "s"(g0), "s"(g1)
                 : "memory");

    // ---------------- Overlap: sum 10 context rows + fetch target row --------
    // float4-chunk mapping: 300 floats = 75 aligned 16B chunks; lane l owns
    // chunks {l, l+32, l+64(<75)}. All accesses become b128 loads.
    const int c0 = lane;
    const int c1 = lane + 32;
    const int c2 = lane + 64;            // valid only for lanes 0..10 (c2 < 75)
    const bool has2 = (c2 < NCHUNK4);

    f32x4 acc0 = {0.f, 0.f, 0.f, 0.f};
    f32x4 acc1 = {0.f, 0.f, 0.f, 0.f};
    f32x4 acc2 = {0.f, 0.f, 0.f, 0.f};

#pragma unroll
    for (int j = 2; j < ILEN; ++j) {
        const f32x4* __restrict__ wr =
            (const f32x4*)(Wg + (size_t)xr[j] * DIM);     // 1200B-aligned base
        acc0 = v4add(acc0, wr[c0]);
        acc1 = v4add(acc1, wr[c1]);
        if (has2) acc2 = v4add(acc2, wr[c2]);
    }

    f32x4 tgt0, tgt1, tgt2 = {0.f, 0.f, 0.f, 0.f};
    {
        const f32x4* __restrict__ wt = (const f32x4*)(Wg + (size_t)idx1 * DIM);
        tgt0 = wt[c0];
        tgt1 = wt[c1];
        if (has2) tgt2 = wt[c2];
    }

    // ---------------- Wait for the sense block, then score ------------------
    __builtin_amdgcn_s_wait_tensorcnt(0);

    float sc[NSENSE];
#pragma unroll
    for (int s = 0; s < NSENSE; ++s) {
        const f32x4* __restrict__ sv =
            (const f32x4*)(&smem[wave][s * DIM]);         // s*1200B: 16B-aligned
        float p = 0.0f;
        p = v4dot(sv[c0], acc0, p);
        p = v4dot(sv[c1], acc1, p);
        if (has2) p = v4dot(sv[c2], acc2, p);
        sc[s] = p;
    }
    // Butterfly reduce: every lane ends with the full 5 scores.
#pragma unroll
    for (int s = 0; s < NSENSE; ++s) {
#pragma unroll
        for (int off = 16; off > 0; off >>= 1)
            sc[s] += __shfl_xor(sc[s], off, 32);
    }

    // argmax with first-max tie-breaking (matches jnp.argmax)
    int best = 0;
    float bv = sc[0];
#pragma unroll
    for (int s = 1; s < NSENSE; ++s) {
        if (sc[s] > bv) { bv = sc[s]; best = s; }
    }

    // ---------------- chosen . W_g[x1] -> sigmoid ----------------------------
    const f32x4* __restrict__ ch = (const f32x4*)(&smem[wave][best * DIM]);
    float p = 0.0f;
    p = v4dot(ch[c0], tgt0, p);
    p = v4dot(ch[c1], tgt1, p);
    if (has2) p = v4dot(ch[c2], tgt2, p);
#pragma unroll
    for (int off = 16; off > 0; off >>= 1)
        p += __shfl_xor(p, off, 32);

    if (lane == 0)
        out[row] = 1.0f / (1.0f + expf(-p));
}

extern "C" void kernel_launch(void* const* d_in, const int* in_sizes, int n_in,
                              void* d_out, int out_size, void* d_ws, size_t ws_size,
                              hipStream_t stream) {
    (void)in_sizes; (void)n_in; (void)d_ws; (void)ws_size; (void)out_size;
    const int*   x  = (const int*)d_in[0];    // (16384, 12) int32
    const float* Wg = (const float*)d_in[1];  // (100000, 300) f32
    const float* Ws = (const float*)d_in[2];  // (100000, 5, 300) f32
    float* out = (float*)d_out;               // (16384,) f32

    dim3 grid(BATCH / WAVES_PER_BLOCK);       // 4096 blocks, 1 wave32 per row
    dim3 block(BLOCK_THREADS);
    SenseEmbedding_kernel<<<grid, block, 0, stream>>>(x, Wg, Ws, out);
}